// TemporalInstantaneousAttention_85899345920593
// MI455X (gfx1250) — compile-verified
//
#include <hip/hip_runtime.h>

// TemporalInstantaneousAttention for MI455X (gfx1250, wave32).
//
// Shapes: B=4, G=8, NH=8, HID=2048, S=4096, dim=256, dk=32, OUT=2048.
// Pipeline: (1) per-(b,g) QKV GEMM 768x256 @ 256xS -> interleaved y (fp32 WMMA)
//           (2) per-(b,head,t) 8x16 sigmoid attention over groups (VALU);
//               og overwrites the q rows of y in place
//           (3) per-(b,g) pred GEMM 256x256 @ 256xS from the og rows (fp32 WMMA)
// S is processed in 512-column chunks so y (~52 MB ws) stays L2-resident
// (192 MB L2). fp32 WMMA (V_WMMA_F32_16X16X4_F32) keeps reference fp32
// numerics while using the matrix pipe.
//
// Each GEMM wave computes 4 M-tiles x 1 N-tile: per K-step, 4x b64 (A) +
// 2x b32 (B) loads feed 4 WMMAs. Out-of-range time columns (the reference's
// zero pad at t==S) are handled branch-free (clamped address * 0/1 mask).
// y is stored INTERLEAVED (row o = 3c + {q,k,v}) so the GEMM epilogue is a
// uniform store loop with no lane-divergent scatter.

#define GQ 8
#define NHQ 8
#define BB 4
#define HIDQ 2048
#define SSQ 4096
#define OUTD 2048
#define DIMQ 256      // HID/G
#define DKQ 32        // DIM/NH
#define OGR 256       // OUT/G
#define M3Q 768       // 3*DIM
#define CHUNK 512
#define KVC (CHUNK + 16)   // y chunk width (need CHUNK+1 cols; round up to 16)
#define MT 4               // M-tiles per wave

typedef __attribute__((ext_vector_type(2))) float v2f;
typedef __attribute__((ext_vector_type(8))) float v8f;

// --------------------------------------------------------------------------
// Kernel 1: QKV GEMM. One wave computes a 64x16 slab (4 WMMA tiles) of
// Y = W(768x256) * X(256xKVC).  X[c,t] = x[b, g*256+c, t] for t<S else 0.
// Stored interleaved: ybuf[bg][o][col], o = 3c+0:q, +1:k, +2:v (bias fused).
// --------------------------------------------------------------------------
__global__ void __launch_bounds__(32)
qkv_gemm_kernel(const float* __restrict__ x, const float* __restrict__ w,
                const float* __restrict__ bias, float* __restrict__ ybuf,
                int s0)
{
    const int lane = threadIdx.x & 31;
    const int half = lane >> 4;          // which 16-lane half of the wave
    const int l16  = lane & 15;
    const int n0   = blockIdx.x * 16;    // 0..527 (33 tiles, covers CHUNK+1)
    const int m0   = blockIdx.y * (16 * MT);   // 0..767 (12 slabs)
    const int bg   = blockIdx.z;         // b*G + g
    const int g    = bg % GQ;

    // Branch-free zero padding for t >= S: clamp address, scale by 0/1 mask.
    const int  tcol = s0 + n0 + l16;
    const int  tc   = (tcol < SSQ) ? tcol : 0;
    const float msk = (tcol < SSQ) ? 1.0f : 0.0f;

    const float* Wg = w + (size_t)g * M3Q * DIMQ;                 // W[o,c]
    const float* Xb = x + ((size_t)(bg / GQ) * HIDQ + g * DIMQ) * SSQ;

    v8f acc[MT] = {};
    for (int k = 0; k < DIMQ; k += 4) {
        const int ka = k + 2 * half;     // lanes<16: K=k..k+1; lanes>=16: K=k+2..k+3
        v2f bvec;
        bvec.x = Xb[(size_t)ka * SSQ + tc] * msk;
        bvec.y = Xb[(size_t)(ka + 1) * SSQ + tc] * msk;
#pragma unroll
        for (int mt = 0; mt < MT; ++mt) {
            v2f a;
            a.x = Wg[(size_t)(m0 + mt * 16 + l16) * DIMQ + ka];
            a.y = Wg[(size_t)(m0 + mt * 16 + l16) * DIMQ + ka + 1];
            acc[mt] = __builtin_amdgcn_wmma_f32_16x16x4_f32(
                /*neg_a=*/false, a, /*neg_b=*/false, bvec,
                /*c_mod=*/(short)0, acc[mt], /*reuse_a=*/false, /*reuse_b=*/false);
        }
    }

    // D layout: VGPR i, lanes 0-15 -> row tile_m0+i, lanes 16-31 -> +8.
    // Uniform epilogue: store interleaved rows directly, no scatter.
    float* Yg = ybuf + (size_t)bg * M3Q * KVC;
    const int n = n0 + l16;
#pragma unroll
    for (int mt = 0; mt < MT; ++mt) {
#pragma unroll
        for (int i = 0; i < 8; ++i) {
            const int m = m0 + mt * 16 + (half << 3) + i;    // row o in [0,768)
            Yg[(size_t)m * KVC + n] = acc[mt][i] + bias[g * M3Q + m];
        }
    }
}

// --------------------------------------------------------------------------
// Kernel 2: instantaneous attention. One thread per (b, head n, group g, t).
// q/k/v live in interleaved ybuf rows 3c / 3c+1 / 3c+2 (c = n*32+d).
// scores[e] = q . k(e, t or t+1) / sqrt(32); z = sigmoid;
// attn = z / (1e-9 + sum z); out = sum_e attn[e]*v[e] overwrites the q rows.
// t is the fastest thread index -> coalesced accesses.
// --------------------------------------------------------------------------
__global__ void __launch_bounds__(256)
attn_kernel(float* __restrict__ ybuf, float* __restrict__ attn_out, int s0)
{
    int t = blockIdx.x * blockDim.x + threadIdx.x;
    const int j = t % CHUNK;  t /= CHUNK;
    const int g = t % GQ;     t /= GQ;
    const int n = t % NHQ;    t /= NHQ;
    const int b = t;
    if (b >= BB) return;

    const float rscale = 0.17677669529663687f;   // 32^-0.5

    float* Yq = ybuf + ((size_t)(b * GQ + g) * M3Q + 3 * (n * DKQ)) * KVC + j;

    float q[DKQ];
#pragma unroll
    for (int d = 0; d < DKQ; ++d) q[d] = Yq[(size_t)(3 * d) * KVC];

    float z[2 * GQ];
    float zsum = 1e-9f;
    for (int e = 0; e < 2 * GQ; ++e) {
        const int eg  = (e < GQ) ? e : (e - GQ);
        const int col = (e < GQ) ? j : (j + 1);      // e>=G uses k at t+1
        const float* Yk = ybuf +
            ((size_t)(b * GQ + eg) * M3Q + 3 * (n * DKQ) + 1) * KVC + col;
        float s = 0.0f;
#pragma unroll
        for (int d = 0; d < DKQ; ++d) s += q[d] * Yk[(size_t)(3 * d) * KVC];
        const float zz = 1.0f / (1.0f + __expf(-s * rscale));
        z[e] = zz;
        zsum += zz;
    }
    const float rz = 1.0f / zsum;

    float outv[DKQ];
#pragma unroll
    for (int d = 0; d < DKQ; ++d) outv[d] = 0.0f;

    const size_t aob = ((((size_t)b * NHQ + n) * GQ + g) * (2 * GQ)) * SSQ + (s0 + j);
    for (int e = 0; e < 2 * GQ; ++e) {
        const float a = z[e] * rz;
        attn_out[aob + (size_t)e * SSQ] = a;         // attn transposed (b,n,g,e,s)
        const int eg  = (e < GQ) ? e : (e - GQ);
        const int col = (e < GQ) ? j : (j + 1);
        const float* Yv = ybuf +
            ((size_t)(b * GQ + eg) * M3Q + 3 * (n * DKQ) + 2) * KVC + col;
#pragma unroll
        for (int d = 0; d < DKQ; ++d) outv[d] += a * Yv[(size_t)(3 * d) * KVC];
    }
    // og overwrites this thread's own q rows (column j only) -> race-free.
#pragma unroll
    for (int d = 0; d < DKQ; ++d) Yq[(size_t)(3 * d) * KVC] = outv[d];
}

// --------------------------------------------------------------------------
// Kernel 3: pred GEMM, one wave per 64x16 slab of Wp(256x256) * og(256xCHUNK).
// og is read from the q rows (3c) of interleaved ybuf; bias fused; stored
// straight to d_out's pred region (B, OUT, S).
// --------------------------------------------------------------------------
__global__ void __launch_bounds__(32)
pred_gemm_kernel(const float* __restrict__ ybuf, const float* __restrict__ w,
                 const float* __restrict__ bias, float* __restrict__ pred,
                 int s0)
{
    const int lane = threadIdx.x & 31;
    const int half = lane >> 4;
    const int l16  = lane & 15;
    const int n0   = blockIdx.x * 16;          // 0..511 (32 tiles)
    const int m0   = blockIdx.y * (16 * MT);   // 0..255 (4 slabs)
    const int bg   = blockIdx.z;
    const int b    = bg / GQ, g = bg % GQ;

    const float* Wg = w + (size_t)g * OGR * DIMQ;          // w_pred[g,o,c]
    const float* Yg = ybuf + (size_t)bg * M3Q * KVC;       // og at rows 3c

    v8f acc[MT] = {};
    for (int k = 0; k < DIMQ; k += 4) {
        const int ka = k + 2 * half;
        v2f bvec;
        bvec.x = Yg[(size_t)(3 * ka) * KVC + n0 + l16];
        bvec.y = Yg[(size_t)(3 * ka + 3) * KVC + n0 + l16];
#pragma unroll
        for (int mt = 0; mt < MT; ++mt) {
            v2f a;
            a.x = Wg[(size_t)(m0 + mt * 16 + l16) * DIMQ + ka];
            a.y = Wg[(size_t)(m0 + mt * 16 + l16) * DIMQ + ka + 1];
            acc[mt] = __builtin_amdgcn_wmma_f32_16x16x4_f32(
                false, a, false, bvec, (short)0, acc[mt], false, false);
        }
    }

    const int n = n0 + l16;
#pragma unroll
    for (int mt = 0; mt < MT; ++mt) {
#pragma unroll
        for (int i = 0; i < 8; ++i) {
            const int m = m0 + mt * 16 + (half << 3) + i;
            pred[((size_t)b * OUTD + g * OGR + m) * SSQ + s0 + n] =
                acc[mt][i] + bias[g * OGR + m];
        }
    }
}

// --------------------------------------------------------------------------
extern "C" void kernel_launch(void* const* d_in, const int* in_sizes, int n_in,
                              void* d_out, int out_size, void* d_ws, size_t ws_size,
                              hipStream_t stream)
{
    (void)in_sizes; (void)n_in; (void)out_size; (void)ws_size;

    const float* x      = (const float*)d_in[0];
    const float* w_qkv  = (const float*)d_in[1];
    const float* b_qkv  = (const float*)d_in[2];
    const float* w_pred = (const float*)d_in[3];
    const float* b_pred = (const float*)d_in[4];

    float* pred     = (float*)d_out;                          // B*OUT*S
    float* attn_out = pred + (size_t)BB * OUTD * SSQ;         // B*NH*G*2G*S

    // Workspace (~52 MB): one interleaved y chunk, (B*G) x 768 x KVC floats.
    float* ybuf = (float*)d_ws;

    for (int s0 = 0; s0 < SSQ; s0 += CHUNK) {
        dim3 g1(KVC / 16, M3Q / (16 * MT), BB * GQ);   // 33 x 12 x 32 waves
        qkv_gemm_kernel<<<g1, 32, 0, stream>>>(x, w_qkv, b_qkv, ybuf, s0);

        const int nthr = BB * NHQ * GQ * CHUNK;        // 131072
        attn_kernel<<<nthr / 256, 256, 0, stream>>>(ybuf, attn_out, s0);

        dim3 g3(CHUNK / 16, OGR / (16 * MT), BB * GQ); // 32 x 4 x 32 waves
        pred_gemm_kernel<<<g3, 32, 0, stream>>>(ybuf, w_pred, b_pred, pred, s0);
    }
}